// ChildSumTreeLSTM_30949534335669
// MI455X (gfx1250) — compile-verified
//
#include <hip/hip_runtime.h>
#include <math.h>

// Child-Sum Tree-LSTM on gfx1250.
// Phase 1: pre = embs @ Wx + bx   via V_WMMA_F32_16X16X4_F32 (fp32 matrix cores).
// Phase 2: depth-level wavefront over the tree (height ~26 for this random
//          increasing tree); each level is a batched 16-row GEMM on WMMA.

#define NN      16384
#define INDIM   300
#define MD      256
#define MAXLEV  64
#define LS      260   // padded LDS row stride (260 % 64 == 4 -> conflict-free col reads)

typedef float v2f __attribute__((ext_vector_type(2)));
typedef float v8f __attribute__((ext_vector_type(8)));

__device__ __forceinline__ float sigmoidf(float x) { return 1.0f / (1.0f + expf(-x)); }

__device__ __forceinline__ v8f wmma4(v2f a, v2f b, v8f c) {
    // D = A(16x4,f32) * B(4x16,f32) + C(16x16,f32)
    return __builtin_amdgcn_wmma_f32_16x16x4_f32(false, a, false, b, (short)0, c, false, false);
}

// ---------------------------------------------------------------- zero scratch
__global__ void k_zero(float* p, size_t n) {
    size_t i  = (size_t)blockIdx.x * blockDim.x + threadIdx.x;
    size_t st = (size_t)gridDim.x * blockDim.x;
    for (; i < n; i += st) p[i] = 0.0f;
}

// ------------------------------------------------- pre = embs @ Wx + bx  (WMMA)
// grid = (NN/16, 2), block = 256 (8 waves). Each wave: one 16x64 output strip.
__global__ __launch_bounds__(256) void k_pre(const float* __restrict__ embs,
                                             const float* __restrict__ Wx,
                                             const float* __restrict__ bx,
                                             float* __restrict__ pre) {
    __shared__ float sA[16 * INDIM];
    const int m0 = blockIdx.x * 16;
    for (int idx = threadIdx.x; idx < 16 * INDIM; idx += 256)
        sA[idx] = embs[(size_t)m0 * INDIM + idx];   // fully coalesced
    __syncthreads();

    const int lane = threadIdx.x & 31, wv = threadIdx.x >> 5;
    const int l15 = lane & 15, hi = lane >> 4;
    const int nb = blockIdx.y * 512 + wv * 64;

    v8f acc[4] = {v8f{}, v8f{}, v8f{}, v8f{}};
    for (int k = 0; k < INDIM; k += 4) {
        const int ka = k + 2 * hi;
        v2f a;
        a.x = sA[l15 * INDIM + ka];
        a.y = sA[l15 * INDIM + ka + 1];
#pragma unroll
        for (int t = 0; t < 4; ++t) {
            const int n = nb + t * 16 + l15;
            v2f b;
            b.x = Wx[(size_t)ka * 1024 + n];
            b.y = Wx[(size_t)(ka + 1) * 1024 + n];
            acc[t] = wmma4(a, b, acc[t]);
        }
    }
#pragma unroll
    for (int t = 0; t < 4; ++t) {
        const int n = nb + t * 16 + l15;
        const float bias = bx[n];
#pragma unroll
        for (int r = 0; r < 8; ++r) {
            const int row = m0 + r + 8 * hi;
            pre[(size_t)row * 1024 + n] = acc[t][r] + bias;
        }
    }
}

// --------------------------------------------- depth relaxation + level buckets
__global__ void k_relax(const int* __restrict__ par, int* depth) {
    int j = blockIdx.x * blockDim.x + threadIdx.x;
    if (j < NN) {
        int d = depth[j];
        atomicMax(&depth[par[j]], d + 1);   // parent[j] > j; values monotone -> converges in <= height passes
    }
}

__global__ void k_hist(const int* __restrict__ depth, int* cnt) {
    int j = blockIdx.x * blockDim.x + threadIdx.x;
    if (j < NN) {
        int d = depth[j]; if (d > MAXLEV - 1) d = MAXLEV - 1;
        atomicAdd(&cnt[d], 1);
    }
}

__global__ void k_scan(const int* __restrict__ cnt, int* off) {
    if (threadIdx.x == 0) {
        int a = 0;
        for (int l = 0; l < MAXLEV; ++l) { off[l] = a; a += cnt[l]; }
        off[MAXLEV] = a;
    }
}

__global__ void k_scatter(const int* __restrict__ depth, const int* __restrict__ off,
                          int* cur, int* __restrict__ order) {
    int j = blockIdx.x * blockDim.x + threadIdx.x;
    if (j < NN) {
        int d = depth[j]; if (d > MAXLEV - 1) d = MAXLEV - 1;
        int pos = off[d] + atomicAdd(&cur[d], 1);
        order[pos] = j;
    }
}

// ----------------------------------------------------- per-level node processing
// block = 512 (16 waves) handles 16 nodes. Wave w owns output columns [16w,16w+16).
// GEMM1: hs(16x256) @ Wh(256x768)  -> three 16x16 tiles (i,o,u) per wave.
// GEMM2: h (16x256) @ Wfh(256x256) -> one 16x16 tile (f) per wave.
__global__ __launch_bounds__(512) void k_level(const float* __restrict__ pre,
                                               const float* __restrict__ Wh,
                                               const float* __restrict__ bh,
                                               const float* __restrict__ Wfh,
                                               const float* __restrict__ bfh,
                                               float* __restrict__ h_sum,
                                               float* __restrict__ fc_sum,
                                               const int* __restrict__ order,
                                               const int* __restrict__ par,
                                               const int* __restrict__ lvl_off,
                                               int lev,
                                               float* __restrict__ h_out) {
    __shared__ int   s_id[16], s_p[16];
    __shared__ float s_hs[16 * LS], s_h[16 * LS], s_c[16 * LS];

    const int s = lvl_off[lev], e = lvl_off[lev + 1];
    const int base = s + (int)blockIdx.x * 16;
    if (base >= e) return;                      // uniform per block
    const int nv = min(16, e - base);

    if (threadIdx.x < 16) {
        int id = NN, p = NN;                    // pad rows point at dummy slot NN
        if ((int)threadIdx.x < nv) { id = order[base + threadIdx.x]; p = par[id]; }
        s_id[threadIdx.x] = id; s_p[threadIdx.x] = p;
    }
    __syncthreads();

    // gather child-h-sums (A matrix for GEMM1)
    for (int idx = threadIdx.x; idx < 16 * MD; idx += 512) {
        int m = idx >> 8, k = idx & 255;
        s_hs[m * LS + k] = h_sum[(size_t)s_id[m] * MD + k];
    }
    __syncthreads();

    const int lane = threadIdx.x & 31, wv = threadIdx.x >> 5;
    const int l15 = lane & 15, hi = lane >> 4;
    const int n0 = wv * 16;

    // ---- GEMM1: i/o/u pre-activations, K = 256
    v8f aI = {}, aO = {}, aU = {};
    for (int k = 0; k < MD; k += 4) {
        const int ka = k + 2 * hi;
        v2f a;
        a.x = s_hs[l15 * LS + ka];
        a.y = s_hs[l15 * LS + ka + 1];
        v2f b;
        b.x = Wh[(size_t)ka * 768 + n0 + l15];
        b.y = Wh[(size_t)(ka + 1) * 768 + n0 + l15];
        aI = wmma4(a, b, aI);
        b.x = Wh[(size_t)ka * 768 + 256 + n0 + l15];
        b.y = Wh[(size_t)(ka + 1) * 768 + 256 + n0 + l15];
        aO = wmma4(a, b, aO);
        b.x = Wh[(size_t)ka * 768 + 512 + n0 + l15];
        b.y = Wh[(size_t)(ka + 1) * 768 + 512 + n0 + l15];
        aU = wmma4(a, b, aU);
    }

    // ---- gates, cell, hidden (elementwise; each lane owns (m,n) pairs)
#pragma unroll
    for (int r = 0; r < 8; ++r) {
        const int m = r + 8 * hi;
        const int n = n0 + l15;
        const int id = s_id[m];
        const float* prow = pre + (size_t)id * 1024;
        float iv = sigmoidf(aI[r] + bh[n]       + prow[n]);
        float ov = sigmoidf(aO[r] + bh[256 + n] + prow[256 + n]);
        float uv = tanhf   (aU[r] + bh[512 + n] + prow[512 + n]);
        float cv = iv * uv + fc_sum[(size_t)id * MD + n];
        float hv = ov * tanhf(cv);
        s_h[m * LS + n] = hv;
        s_c[m * LS + n] = cv;
        if (m < nv) {
            h_out[(size_t)id * MD + n] = hv;
            atomicAdd(&h_sum[(size_t)s_p[m] * MD + n], hv);
        }
    }
    __syncthreads();

    // ---- GEMM2: forget-gate pre-activation f = h @ Wfh, K = 256
    v8f aF = {};
    for (int k = 0; k < MD; k += 4) {
        const int ka = k + 2 * hi;
        v2f a;
        a.x = s_h[l15 * LS + ka];
        a.y = s_h[l15 * LS + ka + 1];
        v2f b;
        b.x = Wfh[(size_t)ka * MD + n0 + l15];
        b.y = Wfh[(size_t)(ka + 1) * MD + n0 + l15];
        aF = wmma4(a, b, aF);
    }
#pragma unroll
    for (int r = 0; r < 8; ++r) {
        const int m = r + 8 * hi;
        const int n = n0 + l15;
        const int p = s_p[m];
        float fx = (p < NN) ? pre[(size_t)p * 1024 + 768 + n] : 0.0f;
        float fv = sigmoidf(aF[r] + bfh[n] + fx);
        if (m < nv) atomicAdd(&fc_sum[(size_t)p * MD + n], fv * s_c[m * LS + n]);
    }
}

// ------------------------------------------------------------------- launcher
extern "C" void kernel_launch(void* const* d_in, const int* in_sizes, int n_in,
                              void* d_out, int out_size, void* d_ws, size_t ws_size,
                              hipStream_t stream) {
    (void)in_sizes; (void)n_in; (void)out_size; (void)ws_size;
    const float* embs   = (const float*)d_in[0];
    const int*   parent = (const int*)  d_in[1];
    const float* Wx     = (const float*)d_in[2];
    const float* bx     = (const float*)d_in[3];
    const float* Wh     = (const float*)d_in[4];
    const float* bh     = (const float*)d_in[5];
    const float* Wfh    = (const float*)d_in[6];
    const float* bfh    = (const float*)d_in[7];
    float* h_out = (float*)d_out;

    // workspace layout (~101 MB)
    float* pre     = (float*)d_ws;                          // NN*1024
    float* h_sum   = pre + (size_t)NN * 1024;               // (NN+1)*MD
    float* fc_sum  = h_sum + (size_t)(NN + 1) * MD;         // (NN+1)*MD
    int*   depth   = (int*)(fc_sum + (size_t)(NN + 1) * MD);// NN+1
    int*   order   = depth + (NN + 1);                      // NN
    int*   lvl_cnt = order + NN;                            // MAXLEV+1
    int*   lvl_off = lvl_cnt + (MAXLEV + 1);                // MAXLEV+1
    int*   cursor  = lvl_off + (MAXLEV + 1);                // MAXLEV

    // zero everything after `pre` (h_sum..cursor) — all 4-byte words
    size_t zero_words = (size_t)(NN + 1) * MD * 2 + (NN + 1) + NN
                        + (MAXLEV + 1) * 2 + MAXLEV;
    k_zero<<<2048, 256, 0, stream>>>(h_sum, zero_words);

    // x-side fused GEMM
    k_pre<<<dim3(NN / 16, 2), 256, 0, stream>>>(embs, Wx, bx, pre);

    // depth = height-from-leaves (monotone relaxation; 64 >> tree height ~26)
    for (int it = 0; it < MAXLEV; ++it)
        k_relax<<<NN / 256, 256, 0, stream>>>(parent, depth);

    // bucket nodes by depth
    k_hist<<<NN / 256, 256, 0, stream>>>(depth, lvl_cnt);
    k_scan<<<1, 32, 0, stream>>>(lvl_cnt, lvl_off);
    k_scatter<<<NN / 256, 256, 0, stream>>>(depth, lvl_off, cursor, order);

    // wavefront over levels; empty blocks self-terminate (graph-capture safe)
    for (int lev = 0; lev < MAXLEV; ++lev)
        k_level<<<NN / 16, 512, 0, stream>>>(pre, Wh, bh, Wfh, bfh,
                                             h_sum, fc_sum, order, parent,
                                             lvl_off, lev, h_out);
}